// ImageWarp_55430847922489
// MI455X (gfx1250) — compile-verified
//
#include <hip/hip_runtime.h>

// Image warp (backproject + project + bilinear sample), MI455X / gfx1250.
// B=64, C=3, H=256, W=512, f32. Memory-bound (~224 MB traffic -> ~10us @ 23.3TB/s).
// Projection is done as a real 16x16x4 f32 WMMA: cp = [Q|t] * (d*x, d*y, d, 1)^T.

#define HDIM 256
#define WDIM 512
#define HWPIX (HDIM * WDIM)   // 131072
#define BATCH 64
#define EPSV 1e-7f

typedef __attribute__((ext_vector_type(2))) float v2f;
typedef __attribute__((ext_vector_type(8))) float v8f;

__global__ __launch_bounds__(256) void ImageWarp_kernel(
    const float* __restrict__ src,   // [B,3,H,W]
    const float* __restrict__ depth, // [B,1,H,W]
    const float* __restrict__ pose,  // [B,4,4]
    const float* __restrict__ Km,    // [B,4,4]
    const float* __restrict__ invK,  // [B,4,4]
    float* __restrict__ out)         // [B,3,H,W]
{
    __shared__ float sA[16][4];      // padded [Q|t] : rows 3..15 zero
    const int b = blockIdx.y;

    if (threadIdx.x == 0) {
        const float* Kb = Km   + b * 16;
        const float* Pb = pose + b * 16;
        const float* Ib = invK + b * 16;
        float P[3][4];
#pragma unroll
        for (int m = 0; m < 3; ++m)
#pragma unroll
            for (int j = 0; j < 4; ++j)
                P[m][j] = Kb[m*4+0]*Pb[0*4+j] + Kb[m*4+1]*Pb[1*4+j]
                        + Kb[m*4+2]*Pb[2*4+j] + Kb[m*4+3]*Pb[3*4+j];
#pragma unroll
        for (int r = 0; r < 16; ++r)
#pragma unroll
            for (int c = 0; c < 4; ++c) sA[r][c] = 0.0f;
#pragma unroll
        for (int m = 0; m < 3; ++m) {
#pragma unroll
            for (int i = 0; i < 3; ++i)
                sA[m][i] = P[m][0]*Ib[0*4+i] + P[m][1]*Ib[1*4+i] + P[m][2]*Ib[2*4+i];
            sA[m][3] = P[m][3];
        }
    }
    __syncthreads();   // full reconvergence: EXEC all-ones for all WMMAs below

    const int lane = threadIdx.x & 31;
    const int wid  = threadIdx.x >> 5;
    const int row  = lane & 15;
    const int cb   = (lane >> 4) << 1;   // 0 for lanes 0-15, 2 for lanes 16-31

    // A-matrix (16x4 f32, 2 VGPRs): lane L holds A[L&15][cb], A[L&15][cb+1]
    v2f a;
    a.x = sA[row][cb];
    a.y = sA[row][cb + 1];

    const float* dep = depth + (size_t)b * HWPIX;
    const float* sb  = src   + (size_t)b * 3 * HWPIX;
    float*       ob  = out   + (size_t)b * 3 * HWPIX;

    const int blockBase = blockIdx.x * 2048;   // 8 waves * 32 px * 8 iters

#pragma unroll 1
    for (int it = 0; it < 8; ++it) {
        const int base = blockBase + it * 256 + wid * 32;  // 32 consecutive px/wave
        const int p    = base + lane;                      // this lane's output pixel

        const float myd = __builtin_nontemporal_load(dep + p);  // streamed once
        const float otd = __shfl_xor(myd, 16, 32);

        // B-matrix (4x16 f32): lanes 0-15 carry rows K=0,1 = (d*x, d*y);
        // lanes 16-31 carry rows K=2,3 = (d, 1). WMMA1 -> pixels base..base+15,
        // WMMA2 -> pixels base+16..base+31.
        const int n  = lane & 15;
        const int pA = base + n;
        const int pB = base + 16 + n;
        const float dA = (lane < 16) ? myd : otd;
        const float dB = (lane < 16) ? otd : myd;
        const float xA = (float)(pA & (WDIM - 1));
        const float yA = (float)(pA >> 9);
        const float xB = (float)(pB & (WDIM - 1));
        const float yB = (float)(pB >> 9);

        v2f b1, b2;
        b1.x = (lane < 16) ? dA * xA : dA;
        b1.y = (lane < 16) ? dA * yA : 1.0f;
        b2.x = (lane < 16) ? dB * xB : dB;
        b2.y = (lane < 16) ? dB * yB : 1.0f;

        v8f c0 = {};
        v8f d1v = __builtin_amdgcn_wmma_f32_16x16x4_f32(
            false, a, false, b1, (short)0, c0, false, false);
        v8f d2v = __builtin_amdgcn_wmma_f32_16x16x4_f32(
            false, a, false, b2, (short)0, c0, false, false);

        // D rows 0..2 live in VGPRs 0..2, lanes 0..15. Redistribute WMMA2's
        // results to lanes 16..31 so the whole wave does the gather phase.
        const float t0 = __shfl_xor(d2v[0], 16, 32);
        const float t1 = __shfl_xor(d2v[1], 16, 32);
        const float t2 = __shfl_xor(d2v[2], 16, 32);
        const float cp0 = (lane < 16) ? d1v[0] : t0;
        const float cp1 = (lane < 16) ? d1v[1] : t1;
        const float cp2 = (lane < 16) ? d1v[2] : t2;

        // perspective divide + grid normalization folded:
        // sample_x = u * W/(W-1) - 0.5 (align_corners=False), border clamp
        const float inv = 1.0f / (cp2 + EPSV);
        float sx = cp0 * inv * ((float)WDIM / (float)(WDIM - 1)) - 0.5f;
        float sy = cp1 * inv * ((float)HDIM / (float)(HDIM - 1)) - 0.5f;
        sx = fminf(fmaxf(sx, 0.0f), (float)(WDIM - 1));
        sy = fminf(fmaxf(sy, 0.0f), (float)(HDIM - 1));

        const float fx0 = floorf(sx);
        const float fy0 = floorf(sy);
        const float wx  = sx - fx0;
        const float wy  = sy - fy0;
        const int ix0 = (int)fx0;
        const int iy0 = (int)fy0;
        const int ix1 = min(ix0 + 1, WDIM - 1);
        const int iy1 = min(iy0 + 1, HDIM - 1);

        const int i00 = iy0 * WDIM + ix0;
        const int i01 = iy0 * WDIM + ix1;
        const int i10 = iy1 * WDIM + ix0;
        const int i11 = iy1 * WDIM + ix1;

        const float w00 = (1.0f - wx) * (1.0f - wy);
        const float w01 = wx * (1.0f - wy);
        const float w10 = (1.0f - wx) * wy;
        const float w11 = wx * wy;

#pragma unroll
        for (int c = 0; c < 3; ++c) {
            const float* pl = sb + c * HWPIX;
            const float v00 = pl[i00];
            const float v01 = pl[i01];
            const float v10 = pl[i10];
            const float v11 = pl[i11];
            // output is write-once: non-temporal store keeps src resident in L2
            __builtin_nontemporal_store(
                v00 * w00 + v01 * w01 + v10 * w10 + v11 * w11,
                ob + c * HWPIX + p);
        }
    }
}

extern "C" void kernel_launch(void* const* d_in, const int* in_sizes, int n_in,
                              void* d_out, int out_size, void* d_ws, size_t ws_size,
                              hipStream_t stream) {
    const float* src   = (const float*)d_in[0];
    const float* depth = (const float*)d_in[1];
    const float* pose  = (const float*)d_in[2];
    const float* Km    = (const float*)d_in[3];
    const float* invK  = (const float*)d_in[4];
    // d_in[5] = cam_type (unused, always 0)
    float* out = (float*)d_out;

    dim3 grid(HWPIX / 2048, BATCH);   // 64 x 64 blocks, 256 threads (8 waves)
    ImageWarp_kernel<<<grid, dim3(256), 0, stream>>>(src, depth, pose, Km, invK, out);
}